// MultiHeadAttention_72816875536681
// MI455X (gfx1250) — compile-verified
//
#include <hip/hip_runtime.h>
#include <math.h>

typedef float v2f __attribute__((ext_vector_type(2)));
typedef float v8f __attribute__((ext_vector_type(8)));

#define DMODEL 512
#define LSEQ   512
#define BATCH  4
#define NHEADF 8    // per-head feature dim (e)
#define DKH    64   // number of "heads" in this module's view
#define MT     64   // query rows per attention block
#define KSTR   12   // padded LDS stride for 8-wide K/V/Q rows (bank-conflict-free)
#define SSTR   516  // padded LDS stride for 512-wide score rows (516%64=4 -> conflict-free)

// ---------------------------------------------------------------------------
// CDNA5 async global->LDS copy (16B per lane), tracked by ASYNCcnt.
// VGLOBAL encoding: vdst = LDS byte offset VGPR, vaddr = 64-bit addr, saddr=off.
// ---------------------------------------------------------------------------
__device__ __forceinline__ void async_copy_b128(void* lds_dst, const void* gsrc) {
    unsigned lofs = (unsigned)(uintptr_t)lds_dst;   // low 32 bits of LDS aperture ptr = LDS offset
    unsigned long long ga = (unsigned long long)(uintptr_t)gsrc;
    asm volatile("global_load_async_to_lds_b128 %0, %1, off"
                 :: "v"(lofs), "v"(ga) : "memory");
}
__device__ __forceinline__ void async_wait_all() {
    asm volatile("s_wait_asynccnt 0x0" ::: "memory");
}

// ---------------------------------------------------------------------------
// LayerNorm over last dim (512) of q. One block (256 threads) per row.
// ---------------------------------------------------------------------------
__global__ __launch_bounds__(256) void ln_kernel(const float* __restrict__ q,
                                                 const float* __restrict__ g,
                                                 const float* __restrict__ beta,
                                                 float* __restrict__ qn) {
    __shared__ float red[256];
    const int row = blockIdx.x;                 // 0 .. B*L-1
    const int t   = threadIdx.x;
    const float* x = q + (size_t)row * DMODEL;
    float v0 = x[t], v1 = x[t + 256];

    red[t] = v0 + v1;
    __syncthreads();
    for (int off = 128; off > 0; off >>= 1) {
        if (t < off) red[t] += red[t + off];
        __syncthreads();
    }
    float mu = red[0] * (1.0f / DMODEL);
    __syncthreads();

    float d0 = v0 - mu, d1 = v1 - mu;
    red[t] = d0 * d0 + d1 * d1;
    __syncthreads();
    for (int off = 128; off > 0; off >>= 1) {
        if (t < off) red[t] += red[t + off];
        __syncthreads();
    }
    float inv = rsqrtf(red[0] * (1.0f / DMODEL) + 1e-6f);

    float* y = qn + (size_t)row * DMODEL;
    y[t]       = d0 * inv * g[t]       + beta[t];
    y[t + 256] = d1 * inv * g[t + 256] + beta[t + 256];
}

// ---------------------------------------------------------------------------
// C[M,N] = A[M,K] @ W[N,K]^T  (+ bias[N]) (+ resid[M,N])
// f32 WMMA 16x16x4. 256 threads = 8 waves; block tile 64x32, wave tile 16x16.
// Weights (1MB) stay resident in 192MB L2; prefetch keeps A streaming ahead.
// ---------------------------------------------------------------------------
__global__ __launch_bounds__(256) void gemm_wmma(const float* __restrict__ A,
                                                 const float* __restrict__ W,
                                                 const float* __restrict__ bias,
                                                 const float* __restrict__ resid,
                                                 float* __restrict__ C,
                                                 int M, int N, int K) {
    const int wave = threadIdx.x >> 5;
    const int lane = threadIdx.x & 31;
    const int half = lane >> 4;       // selects K sub-pair per ISA A/B layout
    const int l16  = lane & 15;

    const int m0 = blockIdx.x * 64 + (wave >> 1) * 16;
    const int n0 = blockIdx.y * 32 + (wave & 1) * 16;

    const float* arow = A + (size_t)(m0 + l16) * K + half * 2;
    const float* brow = W + (size_t)(n0 + l16) * K + half * 2;

    v8f acc = {};
#pragma unroll 4
    for (int k = 0; k < K; k += 4) {
        __builtin_prefetch(arow + k + 64, 0, 1);   // global_prefetch_b8
        v2f a = *(const v2f*)(arow + k);           // A[m][k..k+1] per ISA 16x4 layout
        v2f b = *(const v2f*)(brow + k);           // (W^T)[k..k+1][n] = W[n][k..k+1]
        acc = __builtin_amdgcn_wmma_f32_16x16x4_f32(
            false, a, false, b, (short)0, acc, false, false);
    }

#pragma unroll
    for (int r = 0; r < 8; ++r) {
        int row = m0 + r + half * 8;               // C/D layout: VGPR r -> M=r / r+8
        int col = n0 + l16;
        float v = acc[r];
        if (bias)  v += bias[col];
        if (resid) v += resid[(size_t)row * N + col];
        C[(size_t)row * N + col] = v;
    }
}

// ---------------------------------------------------------------------------
// Fused attention for one (b, dk) head, 64 query rows per block.
//   S = Q K^T / 8 + sim ; P = softmax(S) ; O = P V
// Q/K/V staged into LDS with CDNA5 async global->LDS copies (ASYNCcnt).
// similarity read NT (one-shot 268MB stream), attn written NT (268MB stream).
// ---------------------------------------------------------------------------
__global__ __launch_bounds__(256) void attn_kernel(const float* __restrict__ Qp,
                                                   const float* __restrict__ Kp,
                                                   const float* __restrict__ Vp,
                                                   const float* __restrict__ sim,
                                                   float* __restrict__ attn_out,
                                                   float* __restrict__ Ohead) {
    extern __shared__ float lds[];
    float* sQ   = lds;                       // MT   x KSTR
    float* sK   = sQ   + MT * KSTR;          // LSEQ x KSTR
    float* sV   = sK   + LSEQ * KSTR;        // LSEQ x KSTR
    float* sS   = sV   + LSEQ * KSTR;        // MT   x SSTR
    float* sRed = sS   + MT * SSTR;          // MT x 4 (row max partials)
    float* sSum = sRed + MT * 4;             // MT x 4 (row sum partials)
    float* sInv = sSum + MT * 4;             // MT
    float* sOp  = sInv + MT;                 // 8 waves x 16 x 16 partial O

    const int b  = blockIdx.z;
    const int dk = blockIdx.y;
    const int qt = blockIdx.x;               // 64-row query tile index
    const int tid  = threadIdx.x;
    const int wave = tid >> 5;
    const int lane = tid & 31;
    const int half = lane >> 4;
    const int l16  = lane & 15;

    // ---- async-stage K, V [512 x 8] and Q tile [64 x 8] into LDS ----
    for (int i = tid; i < LSEQ * 2; i += 256) {
        int row = i >> 1, h = (i & 1) * 4;
        size_t gofs = ((size_t)(b * LSEQ + row)) * DMODEL + dk * NHEADF + h;
        async_copy_b128(sK + row * KSTR + h, Kp + gofs);
        async_copy_b128(sV + row * KSTR + h, Vp + gofs);
    }
    for (int i = tid; i < MT * 2; i += 256) {
        int row = i >> 1, h = (i & 1) * 4;
        size_t gofs = ((size_t)(b * LSEQ + qt * MT + row)) * DMODEL + dk * NHEADF + h;
        async_copy_b128(sQ + row * KSTR + h, Qp + gofs);
    }
    async_wait_all();
    __syncthreads();

    // ---- scores: 4x32 = 128 WMMA tiles of 16x16, 16 tiles per wave ----
    const float* simBase =
        sim + ((size_t)(b * DKH + dk) * LSEQ + (size_t)qt * MT) * LSEQ;
#pragma unroll 1
    for (int t = 0; t < 16; ++t) {
        int tile = wave * 16 + t;
        int mt = tile >> 5, nt = tile & 31;
        v8f acc = {};
#pragma unroll
        for (int kb = 0; kb < NHEADF; kb += 4) {
            v2f a = *(const v2f*)(sQ + (mt * 16 + l16) * KSTR + kb + half * 2);
            v2f bb = *(const v2f*)(sK + (nt * 16 + l16) * KSTR + kb + half * 2);
            acc = __builtin_amdgcn_wmma_f32_16x16x4_f32(
                false, a, false, bb, (short)0, acc, false, false);
        }
#pragma unroll
        for (int r = 0; r < 8; ++r) {
            int m = mt * 16 + r + half * 8;
            int n = nt * 16 + l16;
            float sv = __builtin_nontemporal_load(simBase + (size_t)m * LSEQ + n);
            sS[m * SSTR + n] = acc[r] * 0.125f + sv;
        }
    }
    __syncthreads();

    // ---- softmax over 512 cols; 4 threads per row, 128 cols each ----
    {
        int row = tid >> 2, qd = tid & 3;
        float* srow = sS + row * SSTR + qd * 128;
        float mx = -3.402823466e38f;
        for (int i = 0; i < 128; ++i) mx = fmaxf(mx, srow[i]);
        sRed[row * 4 + qd] = mx;
        __syncthreads();
        float rmax = fmaxf(fmaxf(sRed[row * 4 + 0], sRed[row * 4 + 1]),
                           fmaxf(sRed[row * 4 + 2], sRed[row * 4 + 3]));
        float sum = 0.f;
        for (int i = 0; i < 128; ++i) {
            float e = __expf(srow[i] - rmax);
            srow[i] = e;
            sum += e;
        }
        sSum[row * 4 + qd] = sum;
        __syncthreads();
        if (tid < MT)
            sInv[tid] = 1.0f / (sSum[tid * 4 + 0] + sSum[tid * 4 + 1] +
                                sSum[tid * 4 + 2] + sSum[tid * 4 + 3]);
        __syncthreads();
    }

    // ---- normalize in LDS + coalesced non-temporal store of attn ----
    float* attnBase =
        attn_out + ((size_t)(b * DKH + dk) * LSEQ + (size_t)qt * MT) * LSEQ;
    for (int i = tid; i < MT * LSEQ; i += 256) {
        int r = i >> 9, c = i & 511;
        float p = sS[r * SSTR + c] * sInv[r];
        sS[r * SSTR + c] = p;
        __builtin_nontemporal_store(p, attnBase + (size_t)r * LSEQ + c);
    }
    __syncthreads();

    // ---- O = P[64x512] @ V[512x8]; wave pairs split K (256 each), WMMA ----
    {
        int rt = wave >> 1;            // 4 row tiles of 16
        int kh = wave & 1;             // split-K half
        int kb0 = kh * 256;
        v8f acc = {};
#pragma unroll 4
        for (int k = 0; k < 256; k += 4) {
            int kr = kb0 + k + half * 2;
            v2f a = *(const v2f*)(sS + (rt * 16 + l16) * SSTR + kr);
            v2f bb;
            bb.x = (l16 < NHEADF) ? sV[kr * KSTR + l16] : 0.f;
            bb.y = (l16 < NHEADF) ? sV[(kr + 1) * KSTR + l16] : 0.f;
            acc = __builtin_amdgcn_wmma_f32_16x16x4_f32(
                false, a, false, bb, (short)0, acc, false, false);
        }
#pragma unroll
        for (int r = 0; r < 8; ++r)
            sOp[wave * 256 + (r + half * 8) * 16 + l16] = acc[r];
    }
    __syncthreads();

    // ---- combine split-K partials, scatter to Ohead[b, l, dk*8 + nh] ----
    for (int i = tid; i < 4 * 16 * NHEADF; i += 256) {
        int rt = i >> 7, rem = i & 127;
        int m = rem >> 3, n = rem & 7;
        float o = sOp[(rt * 2) * 256 + m * 16 + n] +
                  sOp[(rt * 2 + 1) * 256 + m * 16 + n];
        Ohead[((size_t)(b * LSEQ) + qt * MT + rt * 16 + m) * DMODEL +
              dk * NHEADF + n] = o;
    }
}

// ---------------------------------------------------------------------------
extern "C" void kernel_launch(void* const* d_in, const int* in_sizes, int n_in,
                              void* d_out, int out_size, void* d_ws, size_t ws_size,
                              hipStream_t stream) {
    const float* q    = (const float*)d_in[0];
    const float* k    = (const float*)d_in[1];
    const float* v    = (const float*)d_in[2];
    const float* sim  = (const float*)d_in[3];
    const float* Wq   = (const float*)d_in[4];
    const float* Wk   = (const float*)d_in[5];
    const float* Wv   = (const float*)d_in[6];
    const float* Wfc  = (const float*)d_in[7];
    const float* bfc  = (const float*)d_in[8];
    const float* ln_g = (const float*)d_in[9];
    const float* ln_b = (const float*)d_in[10];

    float* out_main = (float*)d_out;                       // [B, L, DM]
    float* out_attn = (float*)d_out + (size_t)BATCH * LSEQ * DMODEL; // [B,DK,L,L]

    const size_t NEL = (size_t)BATCH * LSEQ * DMODEL;      // 1M floats
    float* qn    = (float*)d_ws;
    float* Qp    = qn + NEL;
    float* Kp    = Qp + NEL;
    float* Vp    = Kp + NEL;
    float* Ohead = Vp + NEL;

    const int M = BATCH * LSEQ;                            // 2048

    // 1) pre-LayerNorm on q
    ln_kernel<<<M, 256, 0, stream>>>(q, ln_g, ln_b, qn);

    // 2) projections (WMMA f32 GEMMs)
    dim3 ggrid(M / 64, DMODEL / 32);
    gemm_wmma<<<ggrid, 256, 0, stream>>>(qn, Wq, nullptr, nullptr, Qp, M, DMODEL, DMODEL);
    gemm_wmma<<<ggrid, 256, 0, stream>>>(k,  Wk, nullptr, nullptr, Kp, M, DMODEL, DMODEL);
    gemm_wmma<<<ggrid, 256, 0, stream>>>(v,  Wv, nullptr, nullptr, Vp, M, DMODEL, DMODEL);

    // 3) fused attention (scores + sim, softmax, attn out, P@V)
    size_t ldsBytes = (size_t)(MT * KSTR + 2 * LSEQ * KSTR + MT * SSTR +
                               MT * 4 * 2 + MT + 8 * 256) * sizeof(float);
    hipFuncSetAttribute((const void*)attn_kernel,
                        hipFuncAttributeMaxDynamicSharedMemorySize, (int)ldsBytes);
    attn_kernel<<<dim3(LSEQ / MT, DKH, BATCH), 256, ldsBytes, stream>>>(
        Qp, Kp, Vp, sim, out_attn, Ohead);

    // 4) output projection + bias + residual
    gemm_wmma<<<ggrid, 256, 0, stream>>>(Ohead, Wfc, bfc, q, out_main, M, DMODEL, DMODEL);
}